// GraphConvolution_5222680232054
// MI455X (gfx1250) — compile-verified
//
#include <hip/hip_runtime.h>

#define N_NODES 100000
#define N_EDGES 1600000
#define D       64          // D_IN == D_OUT == 64

typedef __attribute__((ext_vector_type(2))) float v2f;
typedef __attribute__((ext_vector_type(8))) float v8f;

// ---------------------------------------------------------------------------
// Kernel 1: zero aggregation buffer (aliased with d_out) and degree counts.
// ---------------------------------------------------------------------------
__global__ void gcn_zero_kernel(float* __restrict__ agg,
                                float* __restrict__ counts) {
    int i = blockIdx.x * blockDim.x + threadIdx.x;
    if (i < N_NODES * D) agg[i] = 0.0f;
    if (i < N_NODES)     counts[i] = 0.0f;
}

// ---------------------------------------------------------------------------
// Kernel 2: edge scatter.  16 lanes per edge; each lane loads a float4 of the
// neighbor feature row (fully coalesced: 16 lanes * 16B = one 256B row) and
// does 4 global_atomic_add_f32 into agg[src].  Lane 0 of each edge counts the
// degree.  x (25.6MB) and agg (25.6MB) both live in the 192MB L2, so this is
// L2-atomic bound, which is the best we can do without a sort.
// ---------------------------------------------------------------------------
__global__ void gcn_scatter_kernel(const float* __restrict__ x,
                                   const int*   __restrict__ edge_index,
                                   float* __restrict__ agg,
                                   float* __restrict__ counts) {
    int t = blockIdx.x * blockDim.x + threadIdx.x;
    if (t >= N_EDGES * 16) return;
    const int e    = t >> 4;
    const int part = t & 15;

    const int src = edge_index[e];             // row 0 of edge_index
    const int tgt = edge_index[N_EDGES + e];   // row 1 of edge_index

    const float4 v = *(const float4*)(x + tgt * D + part * 4);
    float* dst = agg + src * D + part * 4;
    atomicAdd(dst + 0, v.x);
    atomicAdd(dst + 1, v.y);
    atomicAdd(dst + 2, v.z);
    atomicAdd(dst + 3, v.w);
    if (part == 0) atomicAdd(counts + src, 1.0f);
}

// ---------------------------------------------------------------------------
// Kernel 3: normalize + GEMM via V_WMMA_F32_16X16X4_F32.
//
// One wave owns a 16(M) x 64(N) output tile.  K = 64 -> 16 WMMA steps of K=4
// per 16x16 N-tile, 4 N-tiles.  A-fragment layout (ISA 7.12.2, 32-bit A 16x4):
//   lanes 0-15  hold (M=lane, K=4k+0 / 4k+1)
//   lanes 16-31 hold (M=lane-16, K=4k+2 / 4k+3)
// Degree normalization is folded into the A fragments.  W (16KB) is staged in
// LDS once per block.  agg aliases the output buffer: each wave reads its
// entire 16x64 tile into registers before writing the same region.
// ---------------------------------------------------------------------------
__global__ void __launch_bounds__(256)
gcn_wmma_kernel(const float* __restrict__ agg,
                const float* __restrict__ counts,
                const float* __restrict__ W,      // [64(in), 64(out)] row-major
                const float* __restrict__ bias,   // [64]
                float* __restrict__ out) {
    __shared__ float Wlds[D * D];
    for (int i = threadIdx.x; i < D * D; i += 256) Wlds[i] = W[i];
    __syncthreads();

    const int lane = threadIdx.x & 31;
    const int wave = threadIdx.x >> 5;
    const int tile = blockIdx.x * 8 + wave;
    const int m0   = tile * 16;
    if (m0 >= N_NODES) return;   // wave-uniform: EXEC all-ones inside WMMA

    const int mrow = lane & 15;
    const int koff = (lane < 16) ? 0 : 2;
    const int row  = m0 + mrow;

    const float invc = 1.0f / (counts[row] + 1e-6f);

    // Preload all 16 normalized A fragments for this wave's 16x64 tile.
    v2f afrag[16];
    const float* arow = agg + row * D;
#pragma unroll
    for (int kk = 0; kk < 16; ++kk) {
        v2f a;
        a.x = arow[kk * 4 + koff];
        a.y = arow[kk * 4 + koff + 1];
        afrag[kk] = a * invc;
    }

    const int mbase = m0 + ((lane < 16) ? 0 : 8);
#pragma unroll
    for (int nt = 0; nt < 4; ++nt) {
        const int n = nt * 16 + mrow;
        v8f acc = {0.f, 0.f, 0.f, 0.f, 0.f, 0.f, 0.f, 0.f};
#pragma unroll
        for (int kk = 0; kk < 16; ++kk) {
            v2f bfrag;   // B (4x16): K in {4kk+koff, 4kk+koff+1}, N = n
            bfrag.x = Wlds[(kk * 4 + koff) * D + n];
            bfrag.y = Wlds[(kk * 4 + koff + 1) * D + n];
            acc = __builtin_amdgcn_wmma_f32_16x16x4_f32(
                /*neg_a=*/false, afrag[kk],
                /*neg_b=*/false, bfrag,
                /*c_mod=*/(short)0, acc,
                /*reuse_a=*/false, /*reuse_b=*/false);
        }
        const float bb = bias[n];
        // D layout: VGPR r -> (M = mbase + r, N = n)
#pragma unroll
        for (int r = 0; r < 8; ++r) {
            out[(mbase + r) * D + n] = acc[r] + bb;
        }
    }
}

// ---------------------------------------------------------------------------
// Launch wrapper.
// d_in order (setup_inputs): x [N*64] f32, edge_index [2*E] i32,
//                            W [64*64] f32, b [64] f32.
// d_out: [N*64] f32.  We alias it as the aggregation buffer (same shape);
// d_ws holds only the 400KB degree-count array.
// ---------------------------------------------------------------------------
extern "C" void kernel_launch(void* const* d_in, const int* in_sizes, int n_in,
                              void* d_out, int out_size, void* d_ws, size_t ws_size,
                              hipStream_t stream) {
    const float* x  = (const float*)d_in[0];
    const int*   ei = (const int*)  d_in[1];
    const float* W  = (const float*)d_in[2];
    const float* b  = (const float*)d_in[3];

    float* out    = (float*)d_out;
    float* agg    = out;            // aliased: agg has identical shape [N,64]
    float* counts = (float*)d_ws;   // N_NODES floats = 400KB

    // 1) zero agg + counts (must run every call: harness poisons buffers)
    {
        int total  = N_NODES * D;
        int blocks = (total + 255) / 256;
        gcn_zero_kernel<<<blocks, 256, 0, stream>>>(agg, counts);
    }
    // 2) scatter-add over edges (16 lanes per edge)
    {
        long long total = (long long)N_EDGES * 16;
        int blocks = (int)((total + 255) / 256);
        gcn_scatter_kernel<<<blocks, 256, 0, stream>>>(x, ei, agg, counts);
    }
    // 3) normalize + GEMM (WMMA f32 16x16x4), 8 tiles of 16 rows per block
    {
        int tiles  = (N_NODES + 15) / 16;           // 6250
        int blocks = (tiles + 7) / 8;               // 782
        gcn_wmma_kernel<<<blocks, 256, 0, stream>>>(agg, counts, W, b, out);
    }
}